// TrajectoryFK_62156766708037
// MI455X (gfx1250) — compile-verified
//
#include <hip/hip_runtime.h>

// Trajectory FK on gfx1250 (MI455X), FP32 throughout (memory-bound: ~200MB
// traffic -> ~9us roofline at 23.3 TB/s). Three-kernel decoupled prefix-scan
// of 3x3 rotation products + per-timestep kinematic tree walk.
// WMMA (v_wmma_f32_16x16x4_f32) batch-applies the uniform block prefix to
// 5 per-lane 3x3 matrices per instruction.

#define SEQ_T   262144
#define BLK     256
#define NB      (SEQ_T / BLK)   // 1024 scan blocks
#define PER_B   (NB / BLK)      // 4 block-products per thread in kernel B

static_assert(NB == BLK * PER_B, "scan sizing");

typedef float v2f __attribute__((ext_vector_type(2)));
typedef float v8f __attribute__((ext_vector_type(8)));

// ---------- small matrix helpers (row-major 3x3) ----------
__device__ __forceinline__ void mat_mul3(const float* __restrict__ A,
                                         const float* __restrict__ B,
                                         float* __restrict__ D) {
#pragma unroll
  for (int i = 0; i < 3; ++i) {
#pragma unroll
    for (int j = 0; j < 3; ++j) {
      D[i*3+j] = fmaf(A[i*3+0], B[0+j], fmaf(A[i*3+1], B[3+j], A[i*3+2]*B[6+j]));
    }
  }
}

__device__ __forceinline__ void mat_vec3(const float* __restrict__ A,
                                         const float* __restrict__ x,
                                         float* __restrict__ y) {
#pragma unroll
  for (int i = 0; i < 3; ++i)
    y[i] = fmaf(A[i*3+0], x[0], fmaf(A[i*3+1], x[1], A[i*3+2]*x[2]));
}

// 6D continuous rotation -> 3x3 matrix; columns are (b1, b2, b1 x b2)
__device__ __forceinline__ void cont6d_to_mat(const float* __restrict__ d6,
                                              float* __restrict__ R) {
  float a1x=d6[0], a1y=d6[1], a1z=d6[2];
  float a2x=d6[3], a2y=d6[4], a2z=d6[5];
  float n1 = sqrtf(fmaf(a1x,a1x, fmaf(a1y,a1y, a1z*a1z)));
  float i1 = 1.0f / fmaxf(n1, 1e-12f);
  float b1x=a1x*i1, b1y=a1y*i1, b1z=a1z*i1;
  float dp = fmaf(b1x,a2x, fmaf(b1y,a2y, b1z*a2z));
  float b2x = fmaf(-dp,b1x,a2x), b2y = fmaf(-dp,b1y,a2y), b2z = fmaf(-dp,b1z,a2z);
  float n2 = sqrtf(fmaf(b2x,b2x, fmaf(b2y,b2y, b2z*b2z)));
  float i2 = 1.0f / fmaxf(n2, 1e-12f);
  b2x*=i2; b2y*=i2; b2z*=i2;
  float b3x = b1y*b2z - b1z*b2y;
  float b3y = b1z*b2x - b1x*b2z;
  float b3z = b1x*b2y - b1y*b2x;
  R[0]=b1x; R[1]=b2x; R[2]=b3x;
  R[3]=b1y; R[4]=b2y; R[5]=b3y;
  R[6]=b1z; R[7]=b2z; R[8]=b3z;
}

// Hillis-Steele inclusive scan of 256 mat3 in LDS, op = (earlier @ later).
// Caller must have stored v into S[tid] and synced. Leaves S[tid] = inclusive.
__device__ __forceinline__ void inclusive_scan_mats(float (*S)[9], int tid, float* v) {
  for (int s = 1; s < BLK; s <<= 1) {
    float L[9];
    const bool has = (tid >= s);
    if (has) {
#pragma unroll
      for (int q = 0; q < 9; ++q) L[q] = S[tid - s][q];
    }
    __syncthreads();
    if (has) {
      float D[9];
      mat_mul3(L, v, D);
#pragma unroll
      for (int q = 0; q < 9; ++q) { v[q] = D[q]; S[tid][q] = D[q]; }
    }
    __syncthreads();
  }
}

// ---------- kernel A: per-block ordered product of joint-0 rotations ----------
__global__ void __launch_bounds__(BLK) fk_block_products(const float* __restrict__ pred,
                                                         float* __restrict__ bp) {
  __shared__ float S[BLK][9];
  const int tid = threadIdx.x;
  const size_t t = (size_t)blockIdx.x * BLK + tid;
  float R[9];
  cont6d_to_mat(pred + t * 126, R);
#pragma unroll
  for (int q = 0; q < 9; ++q) S[tid][q] = R[q];
  __syncthreads();
  for (int s = 1; s < BLK; s <<= 1) {
    if ((tid & (2*s - 1)) == 0) {
      float L[9], Rr[9], D[9];
#pragma unroll
      for (int q = 0; q < 9; ++q) { L[q] = S[tid][q]; Rr[q] = S[tid + s][q]; }
      mat_mul3(L, Rr, D);
#pragma unroll
      for (int q = 0; q < 9; ++q) S[tid][q] = D[q];
    }
    __syncthreads();
  }
  if (tid < 9) bp[blockIdx.x * 9 + tid] = S[0][tid];
}

// ---------- kernel B: exclusive scan of the 1024 block products ----------
__global__ void __launch_bounds__(BLK) fk_scan_block_products(const float* __restrict__ bp,
                                                              float* __restrict__ pre) {
  __shared__ float S[BLK][9];
  const int k = threadIdx.x;
  float T0[9], T1[9], T2[9], v[9], M[9];
#pragma unroll
  for (int q = 0; q < 9; ++q) T0[q] = bp[(k*4 + 0)*9 + q];
#pragma unroll
  for (int q = 0; q < 9; ++q) M[q]  = bp[(k*4 + 1)*9 + q];
  mat_mul3(T0, M, T1);
#pragma unroll
  for (int q = 0; q < 9; ++q) M[q]  = bp[(k*4 + 2)*9 + q];
  mat_mul3(T1, M, T2);
#pragma unroll
  for (int q = 0; q < 9; ++q) M[q]  = bp[(k*4 + 3)*9 + q];
  mat_mul3(T2, M, v);

#pragma unroll
  for (int q = 0; q < 9; ++q) S[k][q] = v[q];
  __syncthreads();
  inclusive_scan_mats(S, k, v);

  float E[9];
  if (k == 0) {
    E[0]=1.f; E[1]=0.f; E[2]=0.f;
    E[3]=0.f; E[4]=1.f; E[5]=0.f;
    E[6]=0.f; E[7]=0.f; E[8]=1.f;
  } else {
#pragma unroll
    for (int q = 0; q < 9; ++q) E[q] = S[k-1][q];
  }
  float O1[9], O2[9], O3[9];
  mat_mul3(E, T0, O1);
  mat_mul3(E, T1, O2);
  mat_mul3(E, T2, O3);
#pragma unroll
  for (int q = 0; q < 9; ++q) {
    pre[(k*4 + 0)*9 + q] = E[q];
    pre[(k*4 + 1)*9 + q] = O1[q];
    pre[(k*4 + 2)*9 + q] = O2[q];
    pre[(k*4 + 3)*9 + q] = O3[q];
  }
}

// ---------- FK tree-walk helpers ----------
__device__ __forceinline__ void fk_step(const float* __restrict__ rp, const float* __restrict__ pp,
                                        const float* __restrict__ row, int j,
                                        const float* __restrict__ offs, int joint,
                                        float* __restrict__ rout, float* __restrict__ pout,
                                        float* __restrict__ o) {
  float m[9];
  cont6d_to_mat(row + j*6, m);
  mat_mul3(rp, m, rout);
  float w[3];
  mat_vec3(rp, offs + joint*3, w);
  pout[0] = pp[0] + w[0]; pout[1] = pp[1] + w[1]; pout[2] = pp[2] + w[2];
  o[joint*3+0] = pout[0]; o[joint*3+1] = pout[1]; o[joint*3+2] = pout[2];
}

__device__ __forceinline__ void fk_leaf(const float* __restrict__ rp, const float* __restrict__ pp,
                                        const float* __restrict__ offs, int joint,
                                        float* __restrict__ o) {
  float w[3];
  mat_vec3(rp, offs + joint*3, w);
  o[joint*3+0] = pp[0] + w[0]; o[joint*3+1] = pp[1] + w[1]; o[joint*3+2] = pp[2] + w[2];
}

// ---------- kernel C: intra-block scan + WMMA prefix-apply + FK ----------
__global__ void __launch_bounds__(BLK) fk_main(const float* __restrict__ pred,
                                               const float* __restrict__ offs,
                                               const float* __restrict__ pre,
                                               float* __restrict__ out) {
  __shared__ float S[BLK][9];
  const int tid = threadIdx.x;
  const size_t t = (size_t)blockIdx.x * BLK + tid;
  const float* row = pred + t * 126;

  float v[9];
  cont6d_to_mat(row, v);
#pragma unroll
  for (int q = 0; q < 9; ++q) S[tid][q] = v[q];
  __syncthreads();
  inclusive_scan_mats(S, tid, v);

  // uniform exclusive block prefix P
  float P[9];
#pragma unroll
  for (int q = 0; q < 9; ++q) P[q] = pre[blockIdx.x * 9 + q];

  // ---- WMMA compose: S[m] <- P @ S[m].  One v_wmma_f32_16x16x4_f32 applies
  // the uniform 3x3 prefix (A rows 0-2, K=3) to FIVE lane matrices packed
  // into B's 16 columns; each wave covers its own 32 LDS slots in 7 ops.
  // Operand builds use branchless selects; LDS loads are unconditional from
  // clamped in-bounds addresses, masked afterwards (avoids exec churn).
  {
    const int lane = tid & 31;
    const int wave = tid >> 5;
    const int hl = lane & 15;       // M (for A) / N (for B,D)
    const bool hi = lane >= 16;     // upper half-wave holds K=2,3 (A) / rows 2,3 (B)

    // A[m][k] = P[m][k] for m,k < 3 else 0  (explicit select trees, no
    // dynamic indexing of the private P[] array)
    const float pa0 = (hl == 0) ? P[0] : (hl == 1) ? P[3] : (hl == 2) ? P[6] : 0.0f; // A[m][0]
    const float pa1 = (hl == 0) ? P[1] : (hl == 1) ? P[4] : (hl == 2) ? P[7] : 0.0f; // A[m][1]
    const float pa2 = (hl == 0) ? P[2] : (hl == 1) ? P[5] : (hl == 2) ? P[8] : 0.0f; // A[m][2]
    v2f a;
    a.x = hi ? pa2 : pa0;           // VGPR0: K=0 (lanes 0-15) / K=2 (lanes 16-31)
    a.y = hi ? 0.0f : pa1;          // VGPR1: K=1 (lanes 0-15) / K=3 (= 0)

#pragma unroll
    for (int g = 0; g < 7; ++g) {
      const int base  = wave*32 + g*5;
      const int cnt   = (g == 6) ? 2 : 5;
      const bool valid = hl < 3*cnt;
      const int mi = min(base + hl/3, BLK - 1);   // clamped: always in-bounds
      const int c  = hl - 3*(hl/3);
      // unconditional LDS loads, then mask
      const float s0 = S[mi][0 + c];              // S_j row0, col c
      const float s1 = S[mi][3 + c];              // S_j row1
      const float s2 = S[mi][6 + c];              // S_j row2
      v2f b;
      b.x = valid ? (hi ? s2 : s0) : 0.0f;        // B[0][n] / B[2][n]
      b.y = (valid && !hi) ? s1 : 0.0f;           // B[1][n] / B[3][n]=0 (dead: A col3=0)
      v8f cz = {0.f,0.f,0.f,0.f,0.f,0.f,0.f,0.f};
      v8f d = __builtin_amdgcn_wmma_f32_16x16x4_f32(
          /*neg_a=*/false, a, /*neg_b=*/false, b,
          /*c_mod=*/(short)0, cz, /*reuse_a=*/false, /*reuse_b=*/false);
      if (!hi && valid) {            // D rows 0..2 live in d[0..2] on lanes 0-15
        S[mi][0 + c] = d[0];
        S[mi][3 + c] = d[1];
        S[mi][6 + c] = d[2];
      }
    }
  }
  __syncthreads();

  // hip[t] = rots[1]
  float r1[9];
#pragma unroll
  for (int q = 0; q < 9; ++q) r1[q] = S[tid][q];

  float* o = out + t * 66;
  o[0] = 0.f; o[1] = 0.f; o[2] = 0.f;                    // pos[0]
  float p1[3] = { offs[3], offs[4], offs[5] };           // pos[1] = offsets[1]
  o[3] = p1[0]; o[4] = p1[1]; o[5] = p1[2];

  float r2[9], ra[9], rb[9];
  float p2[3], pa[3], pb[3];
  // spine: 2,3,4,5, leaf 6
  fk_step(r1, p1, row, 1,  offs, 2,  r2, p2, o);
  fk_step(r2, p2, row, 2,  offs, 3,  ra, pa, o);
  fk_step(ra, pa, row, 3,  offs, 4,  rb, pb, o);
  fk_step(rb, pb, row, 4,  offs, 5,  ra, pa, o);
  fk_leaf(ra, pa, offs, 6, o);
  // chain 7,8,9, leaf 10 (from joint 1)
  fk_step(r1, p1, row, 6,  offs, 7,  ra, pa, o);
  fk_step(ra, pa, row, 7,  offs, 8,  rb, pb, o);
  fk_step(rb, pb, row, 8,  offs, 9,  ra, pa, o);
  fk_leaf(ra, pa, offs, 10, o);
  // chain 11,12,13, leaf 14 (from joint 1)
  fk_step(r1, p1, row, 10, offs, 11, ra, pa, o);
  fk_step(ra, pa, row, 11, offs, 12, rb, pb, o);
  fk_step(rb, pb, row, 12, offs, 13, ra, pa, o);
  fk_leaf(ra, pa, offs, 14, o);
  // chain 15,16, leaf 17 (from joint 2)
  fk_step(r2, p2, row, 14, offs, 15, ra, pa, o);
  fk_step(ra, pa, row, 15, offs, 16, rb, pb, o);
  fk_leaf(rb, pb, offs, 17, o);
  // chain 18,19,20, leaf 21 (from joint 2)
  fk_step(r2, p2, row, 17, offs, 18, ra, pa, o);
  fk_step(ra, pa, row, 18, offs, 19, rb, pb, o);
  fk_step(rb, pb, row, 19, offs, 20, ra, pa, o);
  fk_leaf(ra, pa, offs, 21, o);
}

extern "C" void kernel_launch(void* const* d_in, const int* in_sizes, int n_in,
                              void* d_out, int out_size, void* d_ws, size_t ws_size,
                              hipStream_t stream) {
  (void)in_sizes; (void)n_in; (void)out_size; (void)ws_size;
  const float* pred = (const float*)d_in[0];   // (262144, 21, 6) f32
  const float* offs = (const float*)d_in[1];   // (22, 3) f32
  float* out = (float*)d_out;                  // (262144, 22, 3) f32
  float* ws  = (float*)d_ws;
  float* bp  = ws;            // NB * 9 floats: per-block products
  float* pre = ws + NB * 9;   // NB * 9 floats: exclusive block prefixes

  fk_block_products<<<NB, BLK, 0, stream>>>(pred, bp);
  fk_scan_block_products<<<1, BLK, 0, stream>>>(bp, pre);
  fk_main<<<NB, BLK, 0, stream>>>(pred, offs, pre, out);
}